// VFR_23021024707170
// MI455X (gfx1250) — compile-verified
//
#include <hip/hip_runtime.h>
#include <hip/hip_bf16.h>

typedef __attribute__((ext_vector_type(2))) float v2f;
typedef __attribute__((ext_vector_type(8))) float v8f;

#define C_CH   64
#define KNN_K  16
#define NPTS   24576
#define WSTRIDE 65           // padded LDS stride for W to avoid bank conflicts

// ---------------------------------------------------------------------------
// Pass 1: h[r, o] = sum_c x[r, c] * W[o, c]   (R x 64 GEMM, f32 WMMA)
// Block: 256 threads = 8 waves. Wave (rt, ct) computes rows [rt*16, +16),
// cols [ct*16, +16). Block covers 32 rows x 64 cols.
// ---------------------------------------------------------------------------
__global__ __launch_bounds__(256) void vfr_gemm_kernel(
    const float* __restrict__ x, const float* __restrict__ W,
    float* __restrict__ h)
{
    __shared__ float Wlds[C_CH * WSTRIDE];

    // stage W (64x64 = 16KB) into LDS, padded rows
    for (int i = threadIdx.x; i < C_CH * C_CH; i += 256) {
        int o = i >> 6, c = i & 63;
        Wlds[o * WSTRIDE + c] = W[i];
    }
    __syncthreads();

    const int wave = threadIdx.x >> 5;   // 0..7
    const int lane = threadIdx.x & 31;
    const int rowTile = wave >> 2;       // 0..1
    const int colTile = wave & 3;        // 0..3
    const int rowBase = blockIdx.x * 32 + rowTile * 16;
    const int colBase = colTile * 16;
    const int m  = lane & 15;            // row (A) / col (B) within tile
    const int kh = lane >> 4;            // K-half selector

    const float* xrow = x + (size_t)(rowBase + m) * C_CH;
    const float* wrow = Wlds + (colBase + m) * WSTRIDE;

    v8f acc = {};
#pragma unroll
    for (int kk = 0; kk < 16; ++kk) {
        const int ko = kk * 4 + kh * 2;  // lane holds K = ko, ko+1
        v2f a; a[0] = xrow[ko]; a[1] = xrow[ko + 1];
        v2f b; b[0] = wrow[ko]; b[1] = wrow[ko + 1];
        acc = __builtin_amdgcn_wmma_f32_16x16x4_f32(
            false, a, false, b, (short)0, acc, false, false);
    }

    // D layout: vgpr g -> row (g + 8*kh), col m
    float* out = h + (size_t)rowBase * C_CH + colBase;
#pragma unroll
    for (int g = 0; g < 8; ++g) {
        out[(size_t)(g + 8 * kh) * C_CH + m] = acc[g];
    }
}

// ---------------------------------------------------------------------------
// Pass 2: v[r,c] = max_k ( h[bN + knn[r,k], c] - h[r, c] )
// plus deterministic per-block partial sums / sumsq per channel.
// 16 threads per row (float4 each); 64 rows per block.
// ---------------------------------------------------------------------------
#define ROWS_PER_BLOCK 64

__global__ __launch_bounds__(256) void vfr_edge_kernel(
    const float* __restrict__ h, const int* __restrict__ knn,
    float* __restrict__ v, float* __restrict__ partial, int R)
{
    __shared__ float red[256 * 8];

    const int tid = threadIdx.x;
    const int cg  = tid & 15;            // channel quad id (0..15)
    const int rl  = tid >> 4;            // row slot (0..15)
    const int row0 = blockIdx.x * ROWS_PER_BLOCK;

    float s0 = 0.f, s1 = 0.f, s2 = 0.f, s3 = 0.f;
    float q0 = 0.f, q1 = 0.f, q2 = 0.f, q3 = 0.f;

    for (int rr = rl; rr < ROWS_PER_BLOCK; rr += 16) {
        const int r = row0 + rr;
        if (r < R) {                                   // uniform per 16-group
            const int b = r / NPTS;
            const int rowOffInBatch = b * NPTS;        // neighbor base row

            const float4 hc = *(const float4*)(h + (size_t)r * C_CH + cg * 4);
            // each of the 16 threads fetches one knn index, shared via shfl
            const int kidx = knn[(size_t)r * KNN_K + cg];

            float m0 = -__builtin_huge_valf(), m1 = m0, m2 = m0, m3 = m0;
#pragma unroll
            for (int k = 0; k < KNN_K; ++k) {
                const int idx = __shfl(kidx, k, 16);
                const float4 hn =
                    *(const float4*)(h + (size_t)(rowOffInBatch + idx) * C_CH + cg * 4);
                m0 = fmaxf(m0, hn.x - hc.x);
                m1 = fmaxf(m1, hn.y - hc.y);
                m2 = fmaxf(m2, hn.z - hc.z);
                m3 = fmaxf(m3, hn.w - hc.w);
            }
            float4 o; o.x = m0; o.y = m1; o.z = m2; o.w = m3;
            *(float4*)(v + (size_t)r * C_CH + cg * 4) = o;

            s0 += m0; s1 += m1; s2 += m2; s3 += m3;
            q0 += m0 * m0; q1 += m1 * m1; q2 += m2 * m2; q3 += m3 * m3;
        }
    }

    // fixed-order block reduction -> per-block partials (deterministic)
    float* slot = red + tid * 8;
    slot[0] = s0; slot[1] = s1; slot[2] = s2; slot[3] = s3;
    slot[4] = q0; slot[5] = q1; slot[6] = q2; slot[7] = q3;
    __syncthreads();

    if (tid < C_CH) {                    // tid == channel
        const int g = tid >> 2;          // channel quad
        const int comp = tid & 3;
        float S = 0.f, Q = 0.f;
#pragma unroll
        for (int i = 0; i < 16; ++i) {
            const float* t = red + (g + 16 * i) * 8;
            S += t[comp];
            Q += t[4 + comp];
        }
        float* p = partial + (size_t)blockIdx.x * 128;
        p[tid]       = S;
        p[64 + tid]  = Q;
    }
}

// ---------------------------------------------------------------------------
// Pass 3: reduce partials per channel -> fold BN into scale/bias
// grid = 64 blocks (one per channel)
// ---------------------------------------------------------------------------
__global__ __launch_bounds__(256) void vfr_stats_kernel(
    const float* __restrict__ partial, const float* __restrict__ gamma,
    const float* __restrict__ beta, float* __restrict__ scalebias,
    int nblocks, float invCount)
{
    __shared__ float ls[256], lq[256];
    const int c = blockIdx.x;
    float S = 0.f, Q = 0.f;
    for (int i = threadIdx.x; i < nblocks; i += 256) {
        const float* p = partial + (size_t)i * 128;
        S += p[c];
        Q += p[64 + c];
    }
    ls[threadIdx.x] = S; lq[threadIdx.x] = Q;
    __syncthreads();
    for (int off = 128; off > 0; off >>= 1) {
        if (threadIdx.x < off) {
            ls[threadIdx.x] += ls[threadIdx.x + off];
            lq[threadIdx.x] += lq[threadIdx.x + off];
        }
        __syncthreads();
    }
    if (threadIdx.x == 0) {
        const float mean = ls[0] * invCount;
        const float var  = lq[0] * invCount - mean * mean;
        const float sc   = gamma[c] * rsqrtf(var + 1e-5f);
        scalebias[c]      = sc;
        scalebias[64 + c] = beta[c] - mean * sc;
    }
}

// ---------------------------------------------------------------------------
// Pass 4: out = v * scale[c] + bias[c]  (in place on d_out, float4)
// ---------------------------------------------------------------------------
__global__ __launch_bounds__(256) void vfr_norm_kernel(
    float* __restrict__ v, const float* __restrict__ scalebias, int total4)
{
    __shared__ float sc[C_CH], bs[C_CH];
    if (threadIdx.x < C_CH) {
        sc[threadIdx.x] = scalebias[threadIdx.x];
        bs[threadIdx.x] = scalebias[64 + threadIdx.x];
    }
    __syncthreads();

    const int i = blockIdx.x * 256 + threadIdx.x;
    if (i < total4) {
        float4 d = ((const float4*)v)[i];
        const int c0 = (i & 15) << 2;        // 16 float4 per 64-ch row
        d.x = fmaf(d.x, sc[c0 + 0], bs[c0 + 0]);
        d.y = fmaf(d.y, sc[c0 + 1], bs[c0 + 1]);
        d.z = fmaf(d.z, sc[c0 + 2], bs[c0 + 2]);
        d.w = fmaf(d.w, sc[c0 + 3], bs[c0 + 3]);
        ((float4*)v)[i] = d;
    }
}

// ---------------------------------------------------------------------------
extern "C" void kernel_launch(void* const* d_in, const int* in_sizes, int n_in,
                              void* d_out, int out_size, void* d_ws, size_t ws_size,
                              hipStream_t stream) {
    const float* x     = (const float*)d_in[0];
    const float* W     = (const float*)d_in[1];
    const float* gamma = (const float*)d_in[2];
    const float* beta  = (const float*)d_in[3];
    const int*   knn   = (const int*)d_in[4];
    float* out = (float*)d_out;

    const int Bz = 8;
    const int R  = Bz * NPTS;                 // 196608 rows

    // workspace layout
    float* h        = (float*)d_ws;                       // R*64 floats (50.3 MB)
    const int nb2   = (R + ROWS_PER_BLOCK - 1) / ROWS_PER_BLOCK;   // 3072
    float* partial  = h + (size_t)R * C_CH;               // nb2*128 floats
    float* scalebias = partial + (size_t)nb2 * 128;       // 128 floats

    vfr_gemm_kernel<<<R / 32, 256, 0, stream>>>(x, W, h);
    vfr_edge_kernel<<<nb2, 256, 0, stream>>>(h, knn, out, partial, R);
    vfr_stats_kernel<<<C_CH, 256, 0, stream>>>(partial, gamma, beta, scalebias,
                                               nb2, 1.0f / (float)R);
    const int total4 = R * (C_CH / 4);
    vfr_norm_kernel<<<(total4 + 255) / 256, 256, 0, stream>>>(out, scalebias, total4);
}